// _ProposalLayer_75024488726911
// MI455X (gfx1250) — compile-verified
//
#include <hip/hip_runtime.h>

// ---------------- geometry / constants ----------------
#define NUM_ANCHORS 16
#define FH 256
#define FW 256
#define NTOT (NUM_ANCHORS * FH * FW)   // 1,048,576
#define PRE_TOPN 6000
#define POST_TOPN 300
#define NMS_THRESH 0.7f
#define MIN_SIZE 4.0f
#define CAND_CAP 8192                  // padded pow2 for bitonic sort
#define KEY_NEGINF 0x007FFFFFu         // f2ord(-inf)

typedef float v2f __attribute__((ext_vector_type(2)));
typedef float v8f __attribute__((ext_vector_type(8)));

struct Ctrl {
  unsigned int hist[4][256];
  unsigned int prefix;
  unsigned int rem;
  unsigned int gt_total;
  unsigned int thresh;
  unsigned int cand_cnt;
  int nms_count;
};

// monotonic float<->u32 order-preserving map (larger float => larger key)
__device__ __forceinline__ unsigned int f2ord(float f) {
  unsigned int b = __float_as_uint(f);
  return (b & 0x80000000u) ? ~b : (b | 0x80000000u);
}
__device__ __forceinline__ float ord2f(unsigned int k) {
  unsigned int b = (k & 0x80000000u) ? (k & 0x7FFFFFFFu) : ~k;
  return __uint_as_float(b);
}

// ---------------- init: zero control state + candidate pad ----------------
__global__ void k_init(Ctrl* c, unsigned long long* cand) {
  int t = blockIdx.x * blockDim.x + threadIdx.x;
  if (t < 1024) ((unsigned int*)c->hist)[t] = 0u;
  if (t == 1024) {
    c->prefix = 0u; c->rem = PRE_TOPN; c->gt_total = 0u;
    c->thresh = 0u; c->cand_cnt = 0u; c->nms_count = 0;
  }
  if (t < CAND_CAP) cand[t] = 0ULL;   // pad sorts to the bottom (descending sort)
}

// ---------------- phase 1: decode + filter + msb-histogram (bandwidth bound) ---
__global__ void k_decode(const float* __restrict__ scores,
                         const float* __restrict__ deltas,
                         const float* __restrict__ boxes,
                         float4* __restrict__ props,
                         unsigned int* __restrict__ key,
                         Ctrl* __restrict__ ctrl) {
  __shared__ unsigned int lh[256];
  int t = threadIdx.x;
  lh[t] = 0u;
  __syncthreads();
  int n = blockIdx.x * 256 + t;          // NTOT == 4096*256, exact cover
  // reshape(B,A,H,W,4) is a flat reinterpret -> float4 contiguous per anchor
  float4 d = ((const float4*)deltas)[n];
  float4 b = ((const float4*)boxes)[n];
  float sc = scores[n];
  float a0 = fmaxf(d.x * b.x + b.x, 0.0f);
  float a1 = fmaxf(d.y * b.y + b.y, 0.0f);
  float bw = fmaxf(expf(d.z) * b.z, 0.0f);
  float bh = fmaxf(expf(d.w) * b.w, 0.0f);
  float x2 = fminf(a0 + bw - 1.0f, (float)(FH - 1));
  float y2 = fminf(a1 + bh - 1.0f, (float)(FW - 1));
  float x1 = fminf(a0, (float)(FH - 1));
  float y1 = fminf(a1, (float)(FW - 1));
  float cw = x2 - x1 + 1.0f;
  float ch = y2 - y1 + 1.0f;
  bool keepf = (cw >= MIN_SIZE) && (ch >= MIN_SIZE);
  props[n] = make_float4(x1, y1, cw, ch);
  float m = keepf ? sc : -__builtin_inff();
  unsigned int k = f2ord(m);
  key[n] = k;
  atomicAdd(&lh[k >> 24], 1u);
  __syncthreads();
  if (lh[t]) atomicAdd(&ctrl->hist[0][t], lh[t]);
}

// ---------------- radix-select scan (single thread, 256 bins) ----------------
__global__ void k_scan(Ctrl* c, int L) {
  unsigned int rem = c->rem;
  unsigned int cum = 0u;
  const unsigned int* h = c->hist[L];
  int b = 255;
  for (; b > 0; --b) {
    unsigned int hv = h[b];
    if (cum + hv >= rem) break;
    cum += hv;
  }
  int shift = 24 - 8 * L;
  c->prefix |= ((unsigned int)b) << shift;
  c->gt_total += cum;          // strictly-greater count accumulates across levels
  c->rem = rem - cum;
  if (L == 3) c->thresh = c->prefix;   // exact 6000th-largest key
}

// ---------------- filtered histogram for levels 1..3 ----------------
__global__ void k_hist(const unsigned int* __restrict__ key, Ctrl* __restrict__ c, int L) {
  __shared__ unsigned int lh[256];
  int t = threadIdx.x;
  lh[t] = 0u;
  __syncthreads();
  int n = blockIdx.x * 256 + t;
  unsigned int k = key[n];
  unsigned int mask = 0xFFFFFFFFu << (32 - 8 * L);
  if ((k & mask) == (c->prefix & mask))
    atomicAdd(&lh[(k >> (24 - 8 * L)) & 0xFFu], 1u);
  __syncthreads();
  if (lh[t]) atomicAdd(&c->hist[L][t], lh[t]);
}

// ---------------- compact candidates >= threshold ----------------
// composite (key<<32)|~n : descending sort => score desc, ties index asc (stable top_k)
__global__ void k_compact(const unsigned int* __restrict__ key, Ctrl* __restrict__ c,
                          unsigned long long* __restrict__ cand) {
  int n = blockIdx.x * 256 + threadIdx.x;
  unsigned int k = key[n];
  if (k >= c->thresh) {
    unsigned int pos = atomicAdd(&c->cand_cnt, 1u);
    if (pos < CAND_CAP)
      cand[pos] = ((unsigned long long)k << 32) | (unsigned int)(~n);
  }
}

// ---------------- 8192-wide in-LDS bitonic sort (64KB LDS, one workgroup) -----
__global__ __launch_bounds__(1024) void k_bitonic(unsigned long long* data, int ascending) {
  __shared__ unsigned long long s[CAND_CAP];
  int t = threadIdx.x;
  for (int i = t; i < CAND_CAP; i += 1024) s[i] = data[i];
  __syncthreads();
  for (int k = 2; k <= CAND_CAP; k <<= 1) {
    for (int j = k >> 1; j > 0; j >>= 1) {
      for (int i = t; i < CAND_CAP; i += 1024) {
        int ixj = i ^ j;
        if (ixj > i) {
          bool up = (((i & k) == 0) == (ascending != 0));
          unsigned long long a = s[i], bb = s[ixj];
          bool swap = up ? (a > bb) : (a < bb);
          if (swap) { s[i] = bb; s[ixj] = a; }
        }
      }
      __syncthreads();
    }
  }
  for (int i = t; i < CAND_CAP; i += 1024) data[i] = s[i];
}

// ---------------- gather top-6000, build y2 sort keys ----------------
__global__ void k_prep(const unsigned long long* __restrict__ cand,
                       const float4* __restrict__ props_all,
                       float4* __restrict__ psel, float* __restrict__ area,
                       float* __restrict__ tscore, unsigned int* __restrict__ validA,
                       unsigned long long* __restrict__ comp2) {
  int i = blockIdx.x * 256 + threadIdx.x;
  if (i >= CAND_CAP) return;
  if (i >= PRE_TOPN) { comp2[i] = ~0ULL; return; }     // pad sorts last (ascending)
  unsigned long long cv = cand[i];
  unsigned int k = (unsigned int)(cv >> 32);
  unsigned int n = ~((unsigned int)cv);
  int valid = (cv != 0ULL) && (k != KEY_NEGINF);
  if (!valid) n = 0;
  float4 p = props_all[n];
  psel[i] = p;
  area[i] = p.z * p.w;                 // entry-order areas (reference's quirk)
  tscore[i] = ord2f(k);
  validA[i] = (unsigned int)valid;
  float y2 = p.y + p.w - 1.0f;
  unsigned int yk = valid ? f2ord(y2) : 0xFFFFFFFFu;   // invalid -> sorts last
  comp2[i] = ((unsigned long long)yk << 32) | (unsigned int)i;  // stable ties
}

// ---------------- WMMA-powered AoS->SoA transpose of sorted boxes -------------
// D = A_onehot(16x4) x B_data(4x16): products are x*1 / x*0, exact in f32.
// A layout (ISA 7.12.2): VGPR0 lanes0-15 K=0, lanes16-31 K=2; VGPR1 K=1 / K=3.
// B assumed to mirror (row K striped across lanes; K=0/2 then K=1/3 per VGPR).
// Result D row r (r<4) = field r of boxes 0..15 across lanes -> SoA stores.
__global__ void k_wmma_tr(const unsigned long long* __restrict__ comp2,
                          const float4* __restrict__ psel,
                          const unsigned int* __restrict__ validA,
                          float* __restrict__ sx1, float* __restrict__ sy1,
                          float* __restrict__ sx2, float* __restrict__ sy2,
                          unsigned int* __restrict__ alive,
                          unsigned int* __restrict__ ordArr) {
  int wave = (blockIdx.x * blockDim.x + threadIdx.x) >> 5;
  int lane = threadIdx.x & 31;
  if (wave >= PRE_TOPN / 16) return;    // whole-wave exit: EXEC stays all-ones
  int p = wave * 16 + (lane & 15);
  unsigned int ind = (unsigned int)(comp2[p] & 0xFFFFFFFFu);   // argsort result
  float4 b4 = psel[ind];
  float x1 = b4.x, y1 = b4.y;
  float x2 = b4.x + b4.z - 1.0f, y2 = b4.y + b4.w - 1.0f;
  v2f B; B.x = (lane < 16) ? x1 : x2;   // K=0 : K=2
        B.y = (lane < 16) ? y1 : y2;    // K=1 : K=3
  v2f A; A.x = (lane == 0 || lane == 18) ? 1.0f : 0.0f;  // onehot rows 0..3
        A.y = (lane == 1 || lane == 19) ? 1.0f : 0.0f;
  v8f C = {};
  C = __builtin_amdgcn_wmma_f32_16x16x4_f32(false, A, false, B, (short)0, C,
                                            false, false);
  if (lane < 16) {
    sx1[p] = C[0]; sy1[p] = C[1]; sx2[p] = C[2]; sy2[p] = C[3];
    alive[p] = validA[ind];
    ordArr[p] = ind;
  }
}

// ---------------- greedy NMS (one workgroup; <=300 rounds x 6000 wide) --------
__global__ __launch_bounds__(1024) void k_nms(const float* __restrict__ sx1,
                                              const float* __restrict__ sy1,
                                              const float* __restrict__ sx2,
                                              const float* __restrict__ sy2,
                                              const float* __restrict__ area,
                                              const unsigned int* __restrict__ alive0,
                                              const unsigned int* __restrict__ ordArr,
                                              const float4* __restrict__ psel,
                                              int* __restrict__ keep,
                                              Ctrl* __restrict__ c) {
  __shared__ unsigned char alive[PRE_TOPN];
  __shared__ int red[1024];
  int t = threadIdx.x;
  for (int p = t; p < PRE_TOPN; p += 1024) alive[p] = (unsigned char)alive0[p];
  __syncthreads();
  int count = 0;
  for (int iter = 0; iter < POST_TOPN; ++iter) {
    // find largest alive position (largest y2)
    int best = -1;
    for (int p = t; p < PRE_TOPN; p += 1024)
      if (alive[p]) best = p;          // monotone stride => per-thread max
    red[t] = best;
    __syncthreads();
    for (int s = 512; s > 0; s >>= 1) {
      if (t < s) red[t] = max(red[t], red[t + s]);
      __syncthreads();
    }
    int pos = red[0];
    if (pos < 0) break;                // uniform exit
    unsigned int ind = ordArr[pos];
    float4 pb = psel[ind];             // entry-order chosen box (x1,y1,cw,ch)
    float X1 = pb.x, Y1 = pb.y;
    float X2 = pb.x + pb.z - 1.0f, Y2 = pb.y + pb.w - 1.0f;
    for (int p = t; p < PRE_TOPN; p += 1024) {
      if (!alive[p]) continue;
      float Wi = fmaxf(fminf(sx2[p], X2) - fmaxf(sx1[p], X1) + 1.0f, 0.0f);
      float Hi = fmaxf(fminf(sy2[p], Y2) - fmaxf(sy1[p], Y1) + 1.0f, 0.0f);
      float r = (Wi * Hi) / area[p];   // entry-order area vs sorted-order inter
      if (r >= NMS_THRESH) alive[p] = 0;
    }
    if (t == 0) { alive[pos] = 0; keep[count] = (int)ind; }
    __syncthreads();
    count++;
  }
  if (t == 0) c->nms_count = count;
}

// ---------------- final output gather: 1200 box floats + 300 scores -----------
__global__ void k_output(const Ctrl* __restrict__ c, const int* __restrict__ keep,
                         const float4* __restrict__ psel,
                         const float* __restrict__ tscore, float* __restrict__ out) {
  int s = blockIdx.x * blockDim.x + threadIdx.x;
  if (s >= POST_TOPN) return;
  int cnt = c->nms_count;
  if (s < cnt) {
    int g = keep[s];
    float4 p = psel[g];
    out[s * 4 + 0] = p.x; out[s * 4 + 1] = p.y;
    out[s * 4 + 2] = p.z; out[s * 4 + 3] = p.w;
    out[POST_TOPN * 4 + s] = tscore[g];
  } else {
    out[s * 4 + 0] = 0.0f; out[s * 4 + 1] = 0.0f;
    out[s * 4 + 2] = 0.0f; out[s * 4 + 3] = 0.0f;
    out[POST_TOPN * 4 + s] = 0.0f;
  }
}

// ---------------- workspace layout (bytes, all 16B aligned) ----------------
#define OFF_PROPS  ((size_t)0)                               // NTOT float4 = 16MB
#define OFF_KEY    (OFF_PROPS + (size_t)NTOT * 16)           // NTOT u32 = 4MB
#define OFF_CTRL   (OFF_KEY + (size_t)NTOT * 4)
#define OFF_CAND   (OFF_CTRL + 8192)
#define OFF_COMP2  (OFF_CAND + (size_t)CAND_CAP * 8)
#define OFF_PSEL   (OFF_COMP2 + (size_t)CAND_CAP * 8)
#define OFF_AREA   (OFF_PSEL + (size_t)PRE_TOPN * 16)
#define OFF_TSCORE (OFF_AREA + (size_t)PRE_TOPN * 4)
#define OFF_VALID  (OFF_TSCORE + (size_t)PRE_TOPN * 4)
#define OFF_SX1    (OFF_VALID + (size_t)PRE_TOPN * 4)
#define OFF_SY1    (OFF_SX1 + (size_t)PRE_TOPN * 4)
#define OFF_SX2    (OFF_SY1 + (size_t)PRE_TOPN * 4)
#define OFF_SY2    (OFF_SX2 + (size_t)PRE_TOPN * 4)
#define OFF_ALIVE  (OFF_SY2 + (size_t)PRE_TOPN * 4)
#define OFF_ORD    (OFF_ALIVE + (size_t)PRE_TOPN * 4)
#define OFF_KEEP   (OFF_ORD + (size_t)PRE_TOPN * 4)

extern "C" void kernel_launch(void* const* d_in, const int* in_sizes, int n_in,
                              void* d_out, int out_size, void* d_ws, size_t ws_size,
                              hipStream_t stream) {
  const float* scores = (const float*)d_in[0];
  const float* deltas = (const float*)d_in[1];
  const float* boxes  = (const float*)d_in[2];
  char* ws = (char*)d_ws;

  float4*             props  = (float4*)(ws + OFF_PROPS);
  unsigned int*       key    = (unsigned int*)(ws + OFF_KEY);
  Ctrl*               ctrl   = (Ctrl*)(ws + OFF_CTRL);
  unsigned long long* cand   = (unsigned long long*)(ws + OFF_CAND);
  unsigned long long* comp2  = (unsigned long long*)(ws + OFF_COMP2);
  float4*             psel   = (float4*)(ws + OFF_PSEL);
  float*              area   = (float*)(ws + OFF_AREA);
  float*              tscore = (float*)(ws + OFF_TSCORE);
  unsigned int*       validA = (unsigned int*)(ws + OFF_VALID);
  float*              sx1    = (float*)(ws + OFF_SX1);
  float*              sy1    = (float*)(ws + OFF_SY1);
  float*              sx2    = (float*)(ws + OFF_SX2);
  float*              sy2    = (float*)(ws + OFF_SY2);
  unsigned int*       alive  = (unsigned int*)(ws + OFF_ALIVE);
  unsigned int*       ordArr = (unsigned int*)(ws + OFF_ORD);
  int*                keep   = (int*)(ws + OFF_KEEP);

  const int NBLK = NTOT / 256;   // 4096

  k_init<<<32, 256, 0, stream>>>(ctrl, cand);
  k_decode<<<NBLK, 256, 0, stream>>>(scores, deltas, boxes, props, key, ctrl);
  k_scan<<<1, 1, 0, stream>>>(ctrl, 0);
  for (int L = 1; L <= 3; ++L) {
    k_hist<<<NBLK, 256, 0, stream>>>(key, ctrl, L);
    k_scan<<<1, 1, 0, stream>>>(ctrl, L);
  }
  k_compact<<<NBLK, 256, 0, stream>>>(key, ctrl, cand);
  k_bitonic<<<1, 1024, 0, stream>>>(cand, 0);            // score desc, idx asc
  k_prep<<<CAND_CAP / 256, 256, 0, stream>>>(cand, props, psel, area, tscore,
                                             validA, comp2);
  k_bitonic<<<1, 1024, 0, stream>>>(comp2, 1);           // y2 asc, stable
  k_wmma_tr<<<(PRE_TOPN / 16 + 7) / 8, 256, 0, stream>>>(comp2, psel, validA,
                                                         sx1, sy1, sx2, sy2,
                                                         alive, ordArr);
  k_nms<<<1, 1024, 0, stream>>>(sx1, sy1, sx2, sy2, area, alive, ordArr, psel,
                                keep, ctrl);
  k_output<<<2, 256, 0, stream>>>(ctrl, keep, psel, tscore, (float*)d_out);
}